// PINN_5617817223985
// MI455X (gfx1250) — compile-verified
//
#include <hip/hip_runtime.h>
#include <math.h>

// CDNA5 / gfx1250 PINN Burgers residual via forward 2nd-order jets + f32 WMMA.
//
// Jet per point: (h, h_x, h_t, h_xx), each a 20-vector, propagated through
// 8 tanh layers. Linear part done with V_WMMA_F32_16X16X4_F32:
//   A = 16 points x K(4-chunk) of one jet component (fp32)
//   B = W[k][n] slice, K=4 x N=16 tile (weights transposed+padded in LDS)
//   C/D = 16x16 fp32 accumulator; tanh chain rule applied in-tile with the
//   native gfx1250 V_TANH_F32 transcendental.
//
// Activation rows padded to KP=36 floats: tile-1 columns 20..31 land in the
// pad (computed as exact zeros via zero-padded weights/biases, never read),
// making the whole layer pipeline divergence-free. Stride 36 dwords is
// bank-conflict-free across 16 lanes (gcd(36,64)=4, 16-cycle).

typedef __attribute__((ext_vector_type(2))) float v2f;
typedef __attribute__((ext_vector_type(8))) float v8f;

#define WAVES 4     // waves per block; 16 points per wave
#define HID   20
#define KP    36    // padded activation row stride (floats)
#define NLAY  8

__device__ __forceinline__ v8f wmma_f32x4(v2f a, v2f b, v8f c) {
  // (neg_a, A, neg_b, B, c_mod, C, reuse_a, reuse_b)
  return __builtin_amdgcn_wmma_f32_16x16x4_f32(
      false, a, false, b, (short)0, c, false, false);
}

// gfx1250 native tanh (TRANS unit).
__device__ __forceinline__ float fast_tanh(float x) {
#if __has_builtin(__builtin_amdgcn_tanhf)
  return __builtin_amdgcn_tanhf(x);
#elif __has_builtin(__builtin_amdgcn_tanh_f32)
  return __builtin_amdgcn_tanh_f32(x);
#else
  float r;
  asm("v_tanh_f32 %0, %1\n\tv_nop" : "=v"(r) : "v"(x));
  return r;
#endif
}

// One dense layer step on a 16-point jet batch held in per-wave LDS.
// CHUNKS = number of K=4 chunks (1 for the 2-input layer, 5 for 20-wide).
template <int CHUNKS>
__device__ __forceinline__ void layer_step(const float (*__restrict__ sWtL)[HID],
                                           const float* __restrict__ sBL,
                                           float (*__restrict__ act)[16][KP],
                                           int n0, int khalf) {
  v8f cH0 = {}, cH1 = {}, cX0 = {}, cX1 = {};
  v8f cT0 = {}, cT1 = {}, cS0 = {}, cS1 = {};

#pragma unroll
  for (int c = 0; c < CHUNKS; ++c) {
    const int k0 = 4 * c + 2 * khalf;
    const v2f B0 = *(const v2f*)&sWtL[n0][k0];
    const v2f B1 = *(const v2f*)&sWtL[n0 + 16][k0];
    const v2f Ah = *(const v2f*)&act[0][n0][k0];
    const v2f Ax = *(const v2f*)&act[1][n0][k0];
    const v2f At = *(const v2f*)&act[2][n0][k0];
    const v2f As = *(const v2f*)&act[3][n0][k0];
    cH0 = wmma_f32x4(Ah, B0, cH0);  cH1 = wmma_f32x4(Ah, B1, cH1);
    cX0 = wmma_f32x4(Ax, B0, cX0);  cX1 = wmma_f32x4(Ax, B1, cX1);
    cT0 = wmma_f32x4(At, B0, cT0);  cT1 = wmma_f32x4(At, B1, cT1);
    cS0 = wmma_f32x4(As, B0, cS0);  cS1 = wmma_f32x4(As, B1, cS1);
  }

  const float bv0 = sBL[n0];
  const float bv1 = sBL[n0 + 16];

  // tanh chain rule in accumulator layout, write back point-major.
  // D element (M, N): M = m + 8*khalf, N = n0 + 16*tile. Tile 1 columns
  // >= 20 write zeros into the row pad; no divergence anywhere.
#pragma unroll
  for (int m = 0; m < 8; ++m) {
    const int M = m + 8 * khalf;
    {
      const float z  = cH0[m] + bv0;
      const float a  = fast_tanh(z);
      const float s  = 1.0f - a * a;
      const float zx = cX0[m], zt = cT0[m], zs = cS0[m];
      act[0][M][n0] = a;
      act[1][M][n0] = s * zx;
      act[2][M][n0] = s * zt;
      act[3][M][n0] = s * zs - 2.0f * a * s * zx * zx;
    }
    {
      const float z  = cH1[m] + bv1;
      const float a  = fast_tanh(z);
      const float s  = 1.0f - a * a;
      const float zx = cX1[m], zt = cT1[m], zs = cS1[m];
      const int N = n0 + 16;
      act[0][M][N] = a;
      act[1][M][N] = s * zx;
      act[2][M][N] = s * zt;
      act[3][M][N] = s * zs - 2.0f * a * s * zx * zx;
    }
  }
}

__global__ __launch_bounds__(WAVES * 32)
void pinn_jet_wmma(const float* __restrict__ x, const float* __restrict__ t,
                   const float* __restrict__ W0, const float* __restrict__ b0,
                   const float* __restrict__ W1, const float* __restrict__ b1,
                   const float* __restrict__ W2, const float* __restrict__ b2,
                   const float* __restrict__ W3, const float* __restrict__ b3,
                   const float* __restrict__ W4, const float* __restrict__ b4,
                   const float* __restrict__ W5, const float* __restrict__ b5,
                   const float* __restrict__ W6, const float* __restrict__ b6,
                   const float* __restrict__ W7, const float* __restrict__ b7,
                   const float* __restrict__ Wout, const float* __restrict__ bout,
                   float* __restrict__ out, int n) {
  // Weights transposed: sWt[l][n][k] = W_l[k][n], zero-padded to n<32.
  __shared__ float sWt[NLAY][32][HID];
  __shared__ float sB[NLAY][32];
  __shared__ float sWo[HID];
  __shared__ float sBo;
  // Per-wave jet activations, point-major: [wave][jet 0..3][point 0..15][KP]
  __shared__ float sAct[WAVES][4][16][KP];

  const int tid   = threadIdx.x;
  const int lane  = tid & 31;
  const int wave  = tid >> 5;
  const int n0    = lane & 15;   // column / point index within tile
  const int khalf = lane >> 4;   // 0: K pair {0,1}; 1: K pair {2,3}

  // ---- stage weights into LDS (transposed + padded) ----
  const float* const Wl[NLAY] = {W0, W1, W2, W3, W4, W5, W6, W7};
  const float* const bl[NLAY] = {b0, b1, b2, b3, b4, b5, b6, b7};
#pragma unroll
  for (int l = 0; l < NLAY; ++l) {
    const float* Ws = Wl[l];
    const int fan = (l == 0) ? 2 : HID;
    for (int idx = tid; idx < 32 * HID; idx += WAVES * 32) {
      const int nn = idx / HID;
      const int kk = idx % HID;
      sWt[l][nn][kk] = (nn < HID && kk < fan) ? Ws[kk * HID + nn] : 0.0f;
    }
    if (tid < 32) sB[l][tid] = (tid < HID) ? bl[l][tid] : 0.0f;
  }
  if (tid < HID) sWo[tid] = Wout[tid];
  if (tid == 0) sBo = bout[0];

  // ---- seed jets: h=(x,t), h_x=(1,0), h_t=(0,1), h_xx=0 ----
  const long base = ((long)blockIdx.x * WAVES + wave) * 16;
  {
    float* actp = &sAct[wave][0][0][0];
    for (int i = lane; i < 4 * 16 * KP; i += 32) actp[i] = 0.0f;
  }
  if (lane < 16) {
    const long p = base + lane;
    float xv = 0.0f, tv = 0.0f;
    if (p < n) { xv = x[p]; tv = t[p]; }
    sAct[wave][0][lane][0] = xv;
    sAct[wave][0][lane][1] = tv;
    sAct[wave][1][lane][0] = 1.0f;   // dx seed
    sAct[wave][2][lane][1] = 1.0f;   // dt seed
  }
  __syncthreads();   // weights visible to all waves; seeds are wave-private

  // ---- 8 layers: layer 0 has fan-in 2 (one K-chunk), rest 20 (five) ----
  layer_step<1>(sWt[0], sB[0], sAct[wave], n0, khalf);
#pragma unroll
  for (int l = 1; l < NLAY; ++l) {
    layer_step<5>(sWt[l], sB[l], sAct[wave], n0, khalf);
  }
  // Same-wave LDS is executed in order; slices are wave-private, so no
  // block barriers are needed between layers.

  // ---- output layer (20 -> 1) + residual, lanes 0..15, one point each ----
  if (lane < 16) {
    const long p = base + lane;
    if (p < n) {
      float u = sBo, ux = 0.0f, ut = 0.0f, uxx = 0.0f;
#pragma unroll
      for (int k = 0; k < HID; ++k) {
        const float w = sWo[k];
        u   += sAct[wave][0][lane][k] * w;
        ux  += sAct[wave][1][lane][k] * w;
        ut  += sAct[wave][2][lane][k] * w;
        uxx += sAct[wave][3][lane][k] * w;
      }
      const float NU = 0.0031830988618379067f;  // 0.01/pi
      out[p] = ut + u * ux - NU * uxx;
    }
  }
}

extern "C" void kernel_launch(void* const* d_in, const int* in_sizes, int n_in,
                              void* d_out, int out_size, void* d_ws, size_t ws_size,
                              hipStream_t stream) {
  (void)n_in; (void)d_ws; (void)ws_size; (void)out_size;
  const float* x = (const float*)d_in[0];
  const float* t = (const float*)d_in[1];
  const float* W[NLAY];
  const float* b[NLAY];
  for (int i = 0; i < NLAY; ++i) {
    W[i] = (const float*)d_in[2 + 2 * i];
    b[i] = (const float*)d_in[3 + 2 * i];
  }
  const float* Wout = (const float*)d_in[18];
  const float* bout = (const float*)d_in[19];
  float* out = (float*)d_out;
  const int n = in_sizes[0];

  const int pts_per_block = WAVES * 16;       // 64
  const int grid = (n + pts_per_block - 1) / pts_per_block;
  pinn_jet_wmma<<<grid, WAVES * 32, 0, stream>>>(
      x, t,
      W[0], b[0], W[1], b[1], W[2], b[2], W[3], b[3],
      W[4], b[4], W[5], b[5], W[6], b[6], W[7], b[7],
      Wout, bout, out, n);
}